// GridSamplePScan_7808250544546
// MI455X (gfx1250) — compile-verified
//
#include <hip/hip_runtime.h>
#include <math.h>

// Problem constants (from reference setup_inputs)
#define BB 4
#define LL 24
#define CC 16
#define HH 128
#define WW 128
#define HWX (HH * WW)

typedef __attribute__((ext_vector_type(2))) float v2f;
typedef __attribute__((ext_vector_type(8))) float v8f;

#if defined(__has_builtin)
#if __has_builtin(__builtin_amdgcn_wmma_f32_16x16x4_f32)
#define HAVE_WMMA_F32X4 1
#endif
#endif

// ---------------------------------------------------------------------------
// Kernel A: cum = cumsum(flows, axis=L) expressed as T(24x24 lower-tri ones)
// times F(24 x pixels), computed with exact-fp32 V_WMMA_F32_16X16X4_F32.
// One wave -> one 16(l) x 16(pixel) tile of one (b, comp) slab.
// Waves: B(4) * comp(2) * ltile(2) * ptile(1024) = 16384 -> 2048 blocks x 256.
// ---------------------------------------------------------------------------
#ifdef HAVE_WMMA_F32X4
__global__ __launch_bounds__(256) void cumsum_wmma_kernel(
    const float* __restrict__ flows, float* __restrict__ cum) {
  const int wave = (blockIdx.x << 3) | (threadIdx.x >> 5);
  const int lane = threadIdx.x & 31;
  const int ptile = wave & 1023;
  const int rest  = wave >> 10;          // 0..15
  const int ltile = rest & 1;
  const int comp  = (rest >> 1) & 1;
  const int b     = rest >> 2;

  const int n    = lane & 15;            // A: row M; B/D: column N (pixel)
  const int half = lane >> 4;            // K sub-block selector for A/B
  const int p    = (ptile << 4) + n;     // flat pixel index
  const int lbase = ltile << 4;
  const int lrow  = lbase + n;           // global l for A rows

  // flows[b, k, comp, p] ; k stride = 2*HW
  const float* fbase = flows + ((size_t)(b * LL * 2 + comp)) * HWX + p;

  v8f c = {};  // accumulator starts at 0
#pragma unroll
  for (int kk = 0; kk < LL; kk += 4) {
    const int k0 = kk + (half << 1);     // this lane's K (vgpr0), K+1 (vgpr1)
    v2f a, bv;
    a.x = (k0     <= lrow) ? 1.0f : 0.0f;  // T[lrow, k0]
    a.y = (k0 + 1 <= lrow) ? 1.0f : 0.0f;  // T[lrow, k0+1]
    bv.x = fbase[(size_t)(k0)     * 2 * HWX];
    bv.y = fbase[(size_t)(k0 + 1) * 2 * HWX];
    c = __builtin_amdgcn_wmma_f32_16x16x4_f32(
        /*neg_a=*/false, a, /*neg_b=*/false, bv,
        /*c_mod=*/(short)0, c, /*reuse_a=*/false, /*reuse_b=*/false);
  }

  // D layout: vgpr v, lanes 0-15 -> M=v ; lanes 16-31 -> M=v+8. N = lane&15.
#pragma unroll
  for (int v = 0; v < 8; ++v) {
    const int lg = lbase + v + (half << 3);
    if (lg < LL) {
      cum[((size_t)((b * LL + lg) * 2 + comp)) * HWX + p] = c[v];
    }
  }
}
#endif

// Scalar fallback cumsum (only used if the wmma builtin is unavailable).
__global__ __launch_bounds__(256) void cumsum_scalar_kernel(
    const float* __restrict__ flows, float* __restrict__ cum) {
  int t = blockIdx.x * blockDim.x + threadIdx.x;  // over B*2*HW
  if (t >= BB * 2 * HWX) return;
  const int p = t % HWX;
  const int comp = (t / HWX) & 1;
  const int b = t / (2 * HWX);
  float s = 0.0f;
  for (int k = 0; k < LL; ++k) {
    s += flows[((size_t)((b * LL + k) * 2 + comp)) * HWX + p];
    cum[((size_t)((b * LL + k) * 2 + comp)) * HWX + p] = s;
  }
}

// ---------------------------------------------------------------------------
// Kernel B: h[b,l] = sum_{i<=l} bilinear_warp(images[b,i], cum[l]-cum[i])
// Block = (b, l, 2-row tile of 256 pixels). Thread = pixel.
// x is wrapped (remainder into [0,2)) -> folded into a 2-tap weight remap.
// y is zero-padded -> rows with zero total weight skip all 64 loads (most of
// the work for large l-i, since displacement std is sqrt(l-i)*64 pixels).
// Accumulates all 16 channels in registers; coalesced stores.
// ---------------------------------------------------------------------------
__global__ __launch_bounds__(256) void warp_accum_kernel(
    const float* __restrict__ images, const float* __restrict__ cum,
    float* __restrict__ out) {
  const int tid = threadIdx.x;
  const int x = tid & (WW - 1);
  const int y = (blockIdx.x << 1) | (tid >> 7);
  const int l = blockIdx.y;
  const int b = blockIdx.z;
  const int pix = (y << 7) | x;

  // cum[b, j, comp, pix] = cum[cbase + j*2*HW + comp*HW]
  const size_t cbase = ((size_t)(b * LL * 2)) * HWX + pix;
  const float cLx = cum[cbase + (size_t)l * 2 * HWX];
  const float cLy = cum[cbase + (size_t)l * 2 * HWX + HWX];

  // Exact base grid: -1 + (2k+1)/128 (all exact binary fractions in fp32).
  const float gxb = -1.0f + (float)(2 * x + 1) * (1.0f / 128.0f);
  const float gyb = -1.0f + (float)(2 * y + 1) * (1.0f / 128.0f);

  float acc[CC];
#pragma unroll
  for (int c = 0; c < CC; ++c) acc[c] = 0.0f;

  for (int i = 0; i <= l; ++i) {
    const float dx = cLx - cum[cbase + (size_t)i * 2 * HWX];
    const float dy = cLy - cum[cbase + (size_t)i * 2 * HWX + HWX];

    // x: wrap into [0,2), then to pixel coords in [-0.5, 127.5)
    const float t = gxb + dx + 1.0f;
    const float r = t - 2.0f * floorf(t * 0.5f);
    const float xi = r * 64.0f - 0.5f;
    const float yi = (gyb + dy + 1.0f) * 64.0f - 0.5f;

    const float fx = floorf(xi);
    const float fy = floorf(yi);
    const float wx = xi - fx;
    const float wy = yi - fy;
    const int x0 = (int)fx;
    const int y0 = (int)fy;

    const float vy0 = (y0 >= 0 && y0 < HH) ? (1.0f - wy) : 0.0f;
    const float vy1 = (y0 >= -1 && y0 < HH - 1) ? wy : 0.0f;
    if (vy0 == 0.0f && vy1 == 0.0f) continue;  // sample fully off-screen in y

    const int rc0 = min(max(y0, 0), HH - 1);
    const int rc1 = min(max(y0 + 1, 0), HH - 1);

    // Remap the two x taps onto an always-in-range adjacent pair [xc, xc+1]:
    //   normal x0 in [0,126]: weights (1-wx, wx)
    //   x0 == -1 : only tap x=0 valid (weight wx)    -> load at 0,   (wx, 0)
    //   x0 == 127: only tap x=127 valid (weight 1-wx)-> load at 126, (0, 1-wx)
    int xc;
    float wL, wR;
    if (x0 < 0) {
      xc = 0;        wL = wx;        wR = 0.0f;
    } else if (x0 > WW - 2) {
      xc = WW - 2;   wL = 0.0f;      wR = 1.0f - wx;
    } else {
      xc = x0;       wL = 1.0f - wx; wR = wx;
    }

    const float* img = images + ((size_t)((b * LL + i) * CC)) * HWX;
    const float* p0 = img + rc0 * WW + xc;
    const float* p1 = img + rc1 * WW + xc;
#pragma unroll
    for (int c = 0; c < CC; ++c) {
      const float a00 = p0[(size_t)c * HWX];
      const float a01 = p0[(size_t)c * HWX + 1];
      const float a10 = p1[(size_t)c * HWX];
      const float a11 = p1[(size_t)c * HWX + 1];
      acc[c] += vy0 * (wL * a00 + wR * a01) + vy1 * (wL * a10 + wR * a11);
    }
  }

  const size_t ob = ((size_t)((b * LL + l) * CC)) * HWX + pix;
#pragma unroll
  for (int c = 0; c < CC; ++c) out[ob + (size_t)c * HWX] = acc[c];
}

// ---------------------------------------------------------------------------
extern "C" void kernel_launch(void* const* d_in, const int* in_sizes, int n_in,
                              void* d_out, int out_size, void* d_ws,
                              size_t ws_size, hipStream_t stream) {
  const float* flows = (const float*)d_in[0];   // (B, L, 2, H, W) fp32
  const float* images = (const float*)d_in[1];  // (B, L, C, H, W) fp32
  float* out = (float*)d_out;                   // (B, L, C, H, W) fp32
  float* cum = (float*)d_ws;                    // B*L*2*H*W fp32 = 12.6 MB

#ifdef HAVE_WMMA_F32X4
  // 16384 waves -> 2048 blocks of 8 waves
  cumsum_wmma_kernel<<<2048, 256, 0, stream>>>(flows, cum);
#else
  const int nthr = BB * 2 * HWX;
  cumsum_scalar_kernel<<<(nthr + 255) / 256, 256, 0, stream>>>(flows, cum);
#endif

  dim3 grid(HH / 2, LL, BB);  // (row-pair, l, b)
  warp_accum_kernel<<<grid, 256, 0, stream>>>(images, cum, out);
}